// ClimbGenerator_60765197304488
// MI455X (gfx1250) — compile-verified
//
#include <hip/hip_runtime.h>
#include <hip/hip_bf16.h>

#define N_GRAPHS 5000
#define NPG      198
#define NNODES   (N_GRAPHS * NPG)   // 990000
#define DEG      16
#define NEDGES   (NNODES * DEG)     // 15840000
#define STRIDE   20                 // fixed row stride for node feature buffers

typedef __attribute__((ext_vector_type(2))) float v2f;
typedef __attribute__((ext_vector_type(8))) float v8f;

// ---------------- zero (vectorized) ----------------
__global__ void zero4_kernel(float4* __restrict__ p, int n4) {
    int i = blockIdx.x * blockDim.x + threadIdx.x;
    if (i < n4) {
        float4 z; z.x = 0.f; z.y = 0.f; z.z = 0.f; z.w = 0.f;
        p[i] = z;
    }
}

// ---------------- node embedding: h = x @ W(20x20) + b ----------------
__global__ void embed_kernel(const float* __restrict__ x,
                             const float* __restrict__ W,
                             const float* __restrict__ b,
                             float* __restrict__ h) {
    __shared__ float sW[400];
    __shared__ float sb[20];
    for (int t = threadIdx.x; t < 400; t += blockDim.x) sW[t] = W[t];
    if (threadIdx.x < 20) sb[threadIdx.x] = b[threadIdx.x];
    __syncthreads();
    int i = blockIdx.x * blockDim.x + threadIdx.x;
    if (i >= NNODES) return;
    const float* xp = (const float*)__builtin_assume_aligned(x + (long)i * 20, 16);
    float xi[20];
#pragma unroll
    for (int k = 0; k < 20; ++k) xi[k] = xp[k];
    float* hp = (float*)__builtin_assume_aligned(h + (long)i * STRIDE, 16);
#pragma unroll
    for (int j = 0; j < 20; ++j) {
        float acc = sb[j];
#pragma unroll
        for (int k = 0; k < 20; ++k) acc = fmaf(xi[k], sW[k * 20 + j], acc);
        hp[j] = acc;
    }
}

// ---------------- edge scatter: agg[dst] += w * h[src]  (L2-resident atomics) ----------------
template <int DIN>
__global__ void scatter_kernel(const float* __restrict__ h,
                               float* __restrict__ agg,
                               const int* __restrict__ src,
                               const int* __restrict__ dst,
                               const float* __restrict__ w) {
    int e = blockIdx.x * blockDim.x + threadIdx.x;
    if (e >= NEDGES) return;
    int s = src[e];
    int d = dst[e];
    float wt = w[e];
    const float* hs = (const float*)__builtin_assume_aligned(h + (long)s * STRIDE, 16);
    float* ad = agg + (long)d * STRIDE;
    float m[DIN];
#pragma unroll
    for (int c = 0; c < DIN; ++c) m[c] = wt * hs[c];   // b128-vectorized gather
#pragma unroll
    for (int c = 0; c < DIN; ++c) {
        unsafeAtomicAdd(&ad[c], m[c]);                 // global_atomic_add_f32
    }
}

// ---------------- node update: h = agg@W_rel + b_rel + h@W_root (in-place, opt ReLU) ----------------
template <int DIN, int DOUT, bool RELU>
__global__ void update_kernel(float* __restrict__ h,
                              const float* __restrict__ agg,
                              const float* __restrict__ Wrel,
                              const float* __restrict__ brel,
                              const float* __restrict__ Wroot) {
    __shared__ float sWrel[DIN * DOUT];
    __shared__ float sWroot[DIN * DOUT];
    __shared__ float sb[DOUT];
    for (int t = threadIdx.x; t < DIN * DOUT; t += blockDim.x) {
        sWrel[t]  = Wrel[t];
        sWroot[t] = Wroot[t];
    }
    if (threadIdx.x < DOUT) sb[threadIdx.x] = brel[threadIdx.x];
    __syncthreads();
    int i = blockIdx.x * blockDim.x + threadIdx.x;
    if (i >= NNODES) return;
    const float* ap = (const float*)__builtin_assume_aligned(agg + (long)i * STRIDE, 16);
    float* hp = (float*)__builtin_assume_aligned(h + (long)i * STRIDE, 16);
    float a[DIN], hh[DIN];
#pragma unroll
    for (int k = 0; k < DIN; ++k) { a[k] = ap[k]; hh[k] = hp[k]; }
    float o[DOUT];
#pragma unroll
    for (int j = 0; j < DOUT; ++j) o[j] = sb[j];
#pragma unroll
    for (int k = 0; k < DIN; ++k) {
#pragma unroll
        for (int j = 0; j < DOUT; ++j)
            o[j] = fmaf(a[k], sWrel[k * DOUT + j], fmaf(hh[k], sWroot[k * DOUT + j], o[j]));
    }
#pragma unroll
    for (int j = 0; j < DOUT; ++j) {
        float v = o[j];
        if (RELU) v = v > 0.f ? v : 0.f;
        hp[j] = v;   // safe: each thread writes only its own row prefix
    }
}

// ---------------- FC layers via V_WMMA_F32_16X16X4_F32, one 16x16 tile per wave32 ----------------
// GATHER=true: A[g][k] is the reshape of the conv output: h[(g*198 + k/2)*STRIDE + (k&1)].
// Loads are clamped (not predicated) so the K-loop runs with full EXEC and no divergence;
// only the partial-tile stores are guarded.
template <bool GATHER>
__device__ __forceinline__ v2f load_a_frag(const float* __restrict__ A, int mc, int ka) {
    if (GATHER) {
        // ka is even -> pair (ka, ka+1) = channels 0,1 of node (ka>>1): contiguous 8B
        const v2f* p = (const v2f*)__builtin_assume_aligned(
            A + ((long)mc * NPG + (ka >> 1)) * STRIDE, 8);
        return *p;
    } else {
        const v2f* p = (const v2f*)__builtin_assume_aligned(A + (long)mc * 396 + ka, 8);
        return *p;
    }
}

template <bool GATHER, bool RELU>
__global__ void fc396_wmma_kernel(const float* __restrict__ A,
                                  const float* __restrict__ W,    // [396][396] row-major (K x N)
                                  const float* __restrict__ bias, // [396]
                                  float* __restrict__ out) {      // [5000][396]
    const int K = 396, NO = 396, M = N_GRAPHS;
    const int ntilesN = (NO + 15) / 16;                // 25
    int wavesPerBlock = blockDim.x >> 5;
    int waveId = blockIdx.x * wavesPerBlock + (threadIdx.x >> 5);
    int tileM = waveId / ntilesN;
    int tileN = waveId % ntilesN;
    if (tileM * 16 >= M) return;                       // wave-uniform exit

    int lane = threadIdx.x & 31;
    int half = lane >> 4;                              // 0: K+0,1  1: K+2,3
    int l15  = lane & 15;
    int m = tileM * 16 + l15;                          // A row this lane loads
    int n = tileN * 16 + l15;                          // B col this lane loads
    int mc = m < M ? m : M - 1;                        // clamped load addresses
    int nc = n < NO ? n : NO - 1;

    v8f c;
    float bv = bias[nc];                               // bias folded into accumulator
#pragma unroll
    for (int r = 0; r < 8; ++r) c[r] = bv;

    const float* Wn = W + nc;                          // column base for B loads

    // Software-pipelined K loop: prefetch next fragment before consuming current.
    int ka0 = half * 2;
    v2f a = load_a_frag<GATHER>(A, mc, ka0);
    v2f b;
    b.x = Wn[(long)ka0 * NO];
    b.y = Wn[(long)(ka0 + 1) * NO];
    for (int kk = 4; kk < K; kk += 4) {
        int ka = kk + half * 2;
        v2f an = load_a_frag<GATHER>(A, mc, ka);
        v2f bn;
        bn.x = Wn[(long)ka * NO];
        bn.y = Wn[(long)(ka + 1) * NO];
        // 8 args: (neg_a, A, neg_b, B, c_mod, C, reuse_a, reuse_b)
        c = __builtin_amdgcn_wmma_f32_16x16x4_f32(false, a, false, b, (short)0, c,
                                                  false, false);
        a = an; b = bn;
    }
    c = __builtin_amdgcn_wmma_f32_16x16x4_f32(false, a, false, b, (short)0, c,
                                              false, false);

#pragma unroll
    for (int r = 0; r < 8; ++r) {
        int mo = tileM * 16 + r + half * 8;            // C/D layout: vgpr r -> M=r / M=r+8
        int no = tileN * 16 + l15;
        if (mo < M && no < NO) {
            float v = c[r];
            if (RELU) v = v > 0.f ? v : 0.f;
            out[(long)mo * NO + no] = v;
        }
    }
}

extern "C" void kernel_launch(void* const* d_in, const int* in_sizes, int n_in,
                              void* d_out, int out_size, void* d_ws, size_t ws_size,
                              hipStream_t stream) {
    const float* x    = (const float*)d_in[0];
    const int*   ei   = (const int*)d_in[1];
    const float* ew   = (const float*)d_in[2];
    const float* embW = (const float*)d_in[3];
    const float* embB = (const float*)d_in[4];
    // convs: d_in[5..19] = {W_rel, b_rel, W_root} x 5
    const float* l1W  = (const float*)d_in[20];
    const float* l1b  = (const float*)d_in[21];
    const float* l2W  = (const float*)d_in[22];
    const float* l2b  = (const float*)d_in[23];
    float* out = (float*)d_out;

    float* h   = (float*)d_ws;                           // N x 20 (stride 20), L2-resident
    float* agg = h + (size_t)NNODES * STRIDE;            // N x 20 (stride 20), L2-resident
    float* fc1out = agg;                                 // reuse after convs (5000x396)

    const int* src = ei;
    const int* dst = ei + NEDGES;

    dim3 b256(256);
    int nodeBlocks = (NNODES + 255) / 256;
    int edgeBlocks = (NEDGES + 255) / 256;
    int n4 = NNODES * STRIDE / 4;
    int zeroBlocks = (n4 + 255) / 256;

    embed_kernel<<<nodeBlocks, b256, 0, stream>>>(x, embW, embB, h);

    // layer 0: 20 -> 20, relu
    zero4_kernel<<<zeroBlocks, b256, 0, stream>>>((float4*)agg, n4);
    scatter_kernel<20><<<edgeBlocks, b256, 0, stream>>>(h, agg, src, dst, ew);
    update_kernel<20, 20, true><<<nodeBlocks, b256, 0, stream>>>(
        h, agg, (const float*)d_in[5], (const float*)d_in[6], (const float*)d_in[7]);

    // layer 1: 20 -> 10, relu
    zero4_kernel<<<zeroBlocks, b256, 0, stream>>>((float4*)agg, n4);
    scatter_kernel<20><<<edgeBlocks, b256, 0, stream>>>(h, agg, src, dst, ew);
    update_kernel<20, 10, true><<<nodeBlocks, b256, 0, stream>>>(
        h, agg, (const float*)d_in[8], (const float*)d_in[9], (const float*)d_in[10]);

    // layer 2: 10 -> 10, relu
    zero4_kernel<<<zeroBlocks, b256, 0, stream>>>((float4*)agg, n4);
    scatter_kernel<10><<<edgeBlocks, b256, 0, stream>>>(h, agg, src, dst, ew);
    update_kernel<10, 10, true><<<nodeBlocks, b256, 0, stream>>>(
        h, agg, (const float*)d_in[11], (const float*)d_in[12], (const float*)d_in[13]);

    // layer 3: 10 -> 5, relu
    zero4_kernel<<<zeroBlocks, b256, 0, stream>>>((float4*)agg, n4);
    scatter_kernel<10><<<edgeBlocks, b256, 0, stream>>>(h, agg, src, dst, ew);
    update_kernel<10, 5, true><<<nodeBlocks, b256, 0, stream>>>(
        h, agg, (const float*)d_in[14], (const float*)d_in[15], (const float*)d_in[16]);

    // layer 4: 5 -> 2, no relu
    zero4_kernel<<<zeroBlocks, b256, 0, stream>>>((float4*)agg, n4);
    scatter_kernel<5><<<edgeBlocks, b256, 0, stream>>>(h, agg, src, dst, ew);
    update_kernel<5, 2, false><<<nodeBlocks, b256, 0, stream>>>(
        h, agg, (const float*)d_in[17], (const float*)d_in[18], (const float*)d_in[19]);

    // FC layers (WMMA): 5000x396 @ 396x396
    int ntilesM = (N_GRAPHS + 15) / 16;   // 313
    int ntilesN = (396 + 15) / 16;        // 25
    int totTiles = ntilesM * ntilesN;     // 7825 waves
    int fcBlocks = (totTiles + 7) / 8;    // 8 waves per 256-thread block

    fc396_wmma_kernel<true, true><<<fcBlocks, 256, 0, stream>>>(h, l1W, l1b, fc1out);
    fc396_wmma_kernel<false, false><<<fcBlocks, 256, 0, stream>>>(fc1out, l2W, l2b, out);
}